// DetectionLoss_84731114815694
// MI455X (gfx1250) — compile-verified
//
#include <hip/hip_runtime.h>
#include <hip/hip_bf16.h>
#include <math.h>

#define B_ 32
#define P_ 16384
#define N_ 64
#define S_ 5
#define CHUNKS 64          // blocks per batch in the scan kernel
#define BLK 256            // 8 waves of 32

typedef float v2f __attribute__((ext_vector_type(2)));
typedef float v8f __attribute__((ext_vector_type(8)));

__device__ __forceinline__ float softplus_f(float x) {
    return fmaxf(x, 0.f) + log1pf(expf(-fabsf(x)));
}

// Exact fp32 32-lane sum using V_WMMA_F32_16X16X4_F32:
// A[m][0]=v(lane m), A[m][2]=v(lane 16+m), other K cols 0; B = ones(4x16).
// D[m][n] = v[m] + v[m+16]; lane n holds 8 rows in the v8f accumulator,
// sum those then fold lanes 0-15 with 16-31 via one shuffle.
__device__ __forceinline__ float wave_sum32(float v) {
    v2f a; a.x = v;   a.y = 0.f;
    v2f b; b.x = 1.f; b.y = 1.f;
    v8f c = {0.f,0.f,0.f,0.f,0.f,0.f,0.f,0.f};
    c = __builtin_amdgcn_wmma_f32_16x16x4_f32(false, a, false, b, (short)0, c,
                                              false, false);
    float s = c[0]+c[1]+c[2]+c[3]+c[4]+c[5]+c[6]+c[7];
    s += __shfl_xor(s, 16, 32);
    return s;
}

__global__ void k_init(unsigned long long* best) {
    int i = blockIdx.x * blockDim.x + threadIdx.x;
    if (i < B_ * N_) best[i] = 0ULL;
}

// Pass over all preds: per-target packed max(iou, ~idx) + 4 big sums.
__global__ __launch_bounds__(BLK) void k_scan(
    const float* __restrict__ pscore, const float* __restrict__ pbox,
    const float* __restrict__ pscale, const float* __restrict__ pctx,
    const float* __restrict__ tbox,
    unsigned long long* __restrict__ best, float* __restrict__ part)
{
    __shared__ float tb[N_ * 4];
    __shared__ unsigned long long sbest[N_];
    __shared__ float wsum[8 * 4];

    const int tid = threadIdx.x;
    const int b   = blockIdx.y;
    const int p   = blockIdx.x * BLK + tid;   // always < P_ (64*256)

    tb[tid] = tbox[b * N_ * 4 + tid];         // 256 floats == N_*4
    if (tid < N_) sbest[tid] = 0ULL;
    __syncthreads();

    const float4 pb = ((const float4*)pbox)[(size_t)b * P_ + p];
    const float area_a = (pb.z - pb.x) * (pb.w - pb.y);

    #pragma unroll 4
    for (int n = 0; n < N_; ++n) {
        float tx1 = tb[4*n], ty1 = tb[4*n+1], tx2 = tb[4*n+2], ty2 = tb[4*n+3];
        float w = fminf(pb.z, tx2) - fmaxf(pb.x, tx1);
        float h = fminf(pb.w, ty2) - fmaxf(pb.y, ty1);
        if (w > 0.f && h > 0.f) {
            float inter  = w * h;
            float area_b = (tx2 - tx1) * (ty2 - ty1);
            float iou    = inter / (area_a + area_b - inter);
            unsigned long long pk =
                ((unsigned long long)__float_as_uint(iou) << 32) |
                (unsigned)(~(unsigned)p);      // smaller idx wins ties
            atomicMax(&sbest[n], pk);
        }
    }

    // big per-batch sums (also feed the no-match fallback branch)
    float sc = pscore[(size_t)b * P_ + p];
    float cx = pctx  [(size_t)b * P_ + p];
    float q0 = softplus_f(sc);                                   // conf
    float q1 = fabsf(pb.x)+fabsf(pb.y)+fabsf(pb.z)+fabsf(pb.w);  // box_reg
    float q3 = softplus_f(cx);                                   // ctx_reg
    float q2 = 0.f;                                              // scale_reg
    const float* sr = pscale + ((size_t)b * P_ + p) * S_;
    #pragma unroll
    for (int s = 0; s < S_; ++s) {
        float v = sr[s];
        float a = v + 1e-6f;
        q2 += (a > 0.f) ? v * logf(a) : 0.f;   // log(safe)=0 when safe==1
    }

    // wave32 reductions via WMMA (EXEC is all-ones here)
    float r0 = wave_sum32(q0);
    float r1 = wave_sum32(q1);
    float r2 = wave_sum32(q2);
    float r3 = wave_sum32(q3);
    int lane = tid & 31, wv = tid >> 5;
    if (lane == 0) {
        wsum[wv*4+0] = r0; wsum[wv*4+1] = r1;
        wsum[wv*4+2] = r2; wsum[wv*4+3] = r3;
    }
    __syncthreads();
    if (tid == 0) {     // fixed-order combine -> deterministic
        float s0=0,s1=0,s2=0,s3=0;
        for (int w = 0; w < 8; ++w) {
            s0 += wsum[w*4+0]; s1 += wsum[w*4+1];
            s2 += wsum[w*4+2]; s3 += wsum[w*4+3];
        }
        float* o = part + ((size_t)(b * CHUNKS + blockIdx.x)) * 4;
        o[0]=s0; o[1]=s1; o[2]=s2; o[3]=s3;
    }
    if (tid < N_) {
        unsigned long long v = sbest[tid];
        if (v) atomicMax(&best[b * N_ + tid], v);   // order-independent max
    }
}

// Per batch: matched-anchor losses + deterministic combine of partials.
__global__ __launch_bounds__(64) void k_match(
    const float* __restrict__ pscore, const float* __restrict__ pbox,
    const float* __restrict__ pscale, const float* __restrict__ pctx,
    const float* __restrict__ tbox,  const int* __restrict__ tscale,
    const float* __restrict__ tctx,
    const unsigned long long* __restrict__ best,
    const float* __restrict__ part, float* __restrict__ batch_out)
{
    __shared__ float    iou_s[N_];
    __shared__ unsigned idx_s[N_];
    __shared__ int      val_s[N_];
    __shared__ float cbox[N_], cce[N_], cbce[N_], ccor[N_];

    const int n = threadIdx.x;
    const int b = blockIdx.x;

    unsigned long long pk = best[b * N_ + n];
    float    iou  = __uint_as_float((unsigned)(pk >> 32));
    unsigned idx  = ~(unsigned)(pk & 0xffffffffu);
    int      valid = (iou > 0.5f) ? 1 : 0;
    iou_s[n] = iou; idx_s[n] = idx; val_s[n] = valid;

    float boxc = 0.f, cec = 0.f, bcec = 0.f, corc = 0.f;
    if (valid) {
        const float4 pb = ((const float4*)pbox)[(size_t)b * P_ + idx];
        const float4 tv = ((const float4*)tbox)[b * N_ + n];
        float pc[4] = {pb.x, pb.y, pb.z, pb.w};
        float tc[4] = {tv.x, tv.y, tv.z, tv.w};
        float s = 0.f;
        #pragma unroll
        for (int c = 0; c < 4; ++c) {
            float d = fabsf(pc[c] - tc[c]);
            s += (d < 0.1f) ? (0.5f * d * d / 0.1f) : (d - 0.05f);
        }
        boxc = s * iou;

        const float* sr = pscale + ((size_t)b * P_ + idx) * S_;
        float mx = sr[0];
        #pragma unroll
        for (int c = 1; c < S_; ++c) mx = fmaxf(mx, sr[c]);
        float se = 0.f;
        #pragma unroll
        for (int c = 0; c < S_; ++c) se += expf(sr[c] - mx);
        int t = tscale[b * N_ + n];
        cec = mx + logf(se) - sr[t];                 // -log_softmax[t]

        float x = pctx[(size_t)b * P_ + idx];
        float y = tctx[b * N_ + n];
        bcec = fmaxf(x, 0.f) - x * y + log1pf(expf(-fabsf(x)));
    }
    __syncthreads();
    if (valid) {
        // scatter-max dedupe: count anchor once with its max iou (first on tie)
        int rep = 1;
        for (int m = 0; m < N_; ++m) {
            if (m != n && val_s[m] && idx_s[m] == idx) {
                if (iou_s[m] > iou || (iou_s[m] == iou && m < n)) rep = 0;
            }
        }
        if (rep) corc = pscore[(size_t)b * P_ + idx] * iou;
    }
    cbox[n] = boxc; cce[n] = cec; cbce[n] = bcec; ccor[n] = corc;
    __syncthreads();

    if (n == 0) {
        float bs=0, cs=0, xs=0, rs=0; int nv=0;
        for (int m = 0; m < N_; ++m) {
            bs += cbox[m]; cs += cce[m]; xs += cbce[m]; rs += ccor[m];
            nv += val_s[m];
        }
        float S0=0, S1=0, S2=0, S3=0;
        const float* pp = part + (size_t)b * CHUNKS * 4;
        for (int c = 0; c < CHUNKS; ++c) {
            S0 += pp[c*4+0]; S1 += pp[c*4+1]; S2 += pp[c*4+2]; S3 += pp[c*4+3];
        }
        const float Pf = (float)P_;
        float denom = (float)((nv > 1) ? nv : 1);
        float bl, sl, cl, fl;
        if (nv > 0) {
            bl = bs / (denom * 4.f);
            sl = cs / denom;
            cl = xs / denom;
            fl = (S0 - rs) / Pf;       // bce(x,t) = softplus(x) - x*t
        } else {
            bl = 0.1f * (S1 / (Pf * 4.f));
            sl = 0.1f * (-S2 / (Pf * (float)S_));
            cl = 0.1f * (S3 / Pf);
            fl = S0 / Pf;
        }
        float* o = batch_out + b * 4;
        o[0]=bl; o[1]=sl; o[2]=cl; o[3]=fl;
    }
}

__global__ void k_final(const float* __restrict__ batch_out,
                        float* __restrict__ out) {
    if (threadIdx.x == 0 && blockIdx.x == 0) {
        float bl=0, sl=0, cl=0, fl=0;
        for (int b = 0; b < B_; ++b) {
            bl += batch_out[b*4+0]; sl += batch_out[b*4+1];
            cl += batch_out[b*4+2]; fl += batch_out[b*4+3];
        }
        float inv = 1.f / (float)B_;
        float box = 2.0f * bl * inv;
        float sc  = 1.0f * sl * inv;
        float cx  = 1.5f * cl * inv;
        float cf  = 1.0f * fl * inv;
        out[0] = box + sc + cx + cf;
        out[1] = box; out[2] = sc; out[3] = cx; out[4] = cf;
    }
}

extern "C" void kernel_launch(void* const* d_in, const int* in_sizes, int n_in,
                              void* d_out, int out_size, void* d_ws, size_t ws_size,
                              hipStream_t stream) {
    const float* pscore = (const float*)d_in[0];
    const float* pbox   = (const float*)d_in[1];
    const float* pscale = (const float*)d_in[2];
    const float* pctx   = (const float*)d_in[3];
    const float* tbox   = (const float*)d_in[4];
    const int*   tscale = (const int*)  d_in[5];
    const float* tctx   = (const float*)d_in[6];
    // d_in[7] = target_conf, unused by the reference forward
    float* out = (float*)d_out;

    char* ws = (char*)d_ws;
    unsigned long long* best = (unsigned long long*)ws;                 // 16 KB
    float* part      = (float*)(ws + (size_t)B_ * N_ * 8);              // 32 KB
    float* batch_out = (float*)(ws + (size_t)B_ * N_ * 8
                                   + (size_t)B_ * CHUNKS * 4 * 4);      // 512 B

    k_init<<<(B_ * N_ + 255) / 256, 256, 0, stream>>>(best);
    dim3 grid(CHUNKS, B_);
    k_scan<<<grid, BLK, 0, stream>>>(pscore, pbox, pscale, pctx, tbox,
                                     best, part);
    k_match<<<B_, N_, 0, stream>>>(pscore, pbox, pscale, pctx, tbox, tscale,
                                   tctx, best, part, batch_out);
    k_final<<<1, 32, 0, stream>>>(batch_out, out);
}